// BiRWKVLLADAModel_10222022165143
// MI455X (gfx1250) — compile-verified
//
#include <hip/hip_runtime.h>
#include <cstdint>

// Problem constants (from reference)
#define BB 8
#define TT 2048
#define CC 2048
#define TD 512

// Tiling
#define NCHUNK 32               // chunks along T
#define LCH (TT / NCHUNK)       // 64 timesteps per chunk
#define VEC 4                   // channels per lane (float4)
#define BLOCK 256
#define CPB (BLOCK * VEC)       // 1024 channels per block
#define TSTEP 4                 // timesteps staged per LDS tile
#define NTILES (LCH / TSTEP)    // 16 tiles per chunk

// ---------------------------------------------------------------------------
// Small helpers
// ---------------------------------------------------------------------------
__device__ __forceinline__ float4 ld4(const float* p) { return *(const float4*)p; }
__device__ __forceinline__ void splat(const float4 q, float* o) {
    o[0] = q.x; o[1] = q.y; o[2] = q.z; o[3] = q.w;
}
__device__ __forceinline__ float4 pack(const float* a) {
    return make_float4(a[0], a[1], a[2], a[3]);
}

// Generic pointer to __shared__ memory: low 32 bits are the wave-relative LDS
// byte offset (generic LDS addr = {SHARED_BASE, offset}).
__device__ __forceinline__ uint32_t lds_addr(const void* p) {
    return (uint32_t)(size_t)p;
}

// CDNA5 async global->LDS copy (16B per lane). Tracked with ASYNCcnt.
__device__ __forceinline__ void async_copy16(uint32_t lds, const float* g) {
    asm volatile("global_load_async_to_lds_b128 %0, %1, off"
                 :: "v"(lds), "v"((unsigned long long)(size_t)g)
                 : "memory");
}
__device__ __forceinline__ void wait_async_le8() {
    asm volatile("s_wait_asynccnt 0x8" ::: "memory");
}
__device__ __forceinline__ void wait_async_0() {
    asm volatile("s_wait_asynccnt 0x0" ::: "memory");
}

// ---------------------------------------------------------------------------
// P0: tf[b] = sigmoid(sum(time_emb[b,:]))
// ---------------------------------------------------------------------------
__global__ __launch_bounds__(256) void BiRWKV_timefac(const float* __restrict__ te,
                                                      float* __restrict__ tf) {
    __shared__ float red[256];
    const int b = blockIdx.x;
    float s = te[(size_t)b * TD + threadIdx.x] + te[(size_t)b * TD + threadIdx.x + 256];
    red[threadIdx.x] = s;
    __syncthreads();
    for (int w = 128; w > 0; w >>= 1) {
        if (threadIdx.x < (unsigned)w) red[threadIdx.x] += red[threadIdx.x + w];
        __syncthreads();
    }
    if (threadIdx.x == 0) tf[b] = 1.0f / (1.0f + __expf(-red[0]));
}

// ---------------------------------------------------------------------------
// P1: forward local scan per chunk; also accumulates the chunk's BACKWARD
// aggregate as a decay-weighted sum (weight decay^(t-t0), ascending t).
// Stores nf_local/df_local per element, plus 4 chunk aggregates.
// k/v are streamed through LDS with double-buffered async-to-LDS loads.
// ---------------------------------------------------------------------------
__global__ __launch_bounds__(BLOCK) void
BiRWKV_fwd_scan(const float* __restrict__ k, const float* __restrict__ v,
                const float* __restrict__ w, const float* __restrict__ tf,
                float* __restrict__ nfl, float* __restrict__ dfl,
                float* __restrict__ aggFa, float* __restrict__ aggFb,
                float* __restrict__ aggBa, float* __restrict__ aggBb) {
    __shared__ __align__(16) float sk[2][TSTEP][CPB];
    __shared__ __align__(16) float sv[2][TSTEP][CPB];
    const int tid = threadIdx.x;
    const int c0  = blockIdx.x * CPB + tid * VEC;
    const int ch  = blockIdx.y;
    const int b   = blockIdx.z;
    const int t0  = ch * LCH;
    const size_t base = ((size_t)b * TT + t0) * (size_t)CC + c0;

    const float sfac = 0.5f + 0.5f * tf[b];
    float wv[VEC];
    splat(ld4(w + c0), wv);
    float dec[VEC];
#pragma unroll
    for (int i = 0; i < VEC; ++i) dec[i] = __expf(-__expf(wv[i])) * sfac;

    const uint32_t lk0 = lds_addr(&sk[0][0][tid * VEC]);
    const uint32_t lv0 = lds_addr(&sv[0][0][tid * VEC]);

    auto prefetch = [&](int it, int buf) {
#pragma unroll
        for (int tt = 0; tt < TSTEP; ++tt) {
            const size_t  o  = base + (size_t)(it * TSTEP + tt) * CC;
            const uint32_t lo = (uint32_t)((buf * TSTEP + tt) * CPB * 4);
            async_copy16(lk0 + lo, k + o);
            async_copy16(lv0 + lo, v + o);
        }
    };

    float aF[VEC] = {0.f, 0.f, 0.f, 0.f}, bF[VEC] = {0.f, 0.f, 0.f, 0.f};
    float aB[VEC] = {0.f, 0.f, 0.f, 0.f}, bB[VEC] = {0.f, 0.f, 0.f, 0.f};
    float pw[VEC] = {1.f, 1.f, 1.f, 1.f};

    prefetch(0, 0);
    for (int it = 0; it < NTILES; ++it) {
        const int buf = it & 1;
        if (it + 1 < NTILES) { prefetch(it + 1, buf ^ 1); wait_async_le8(); }
        else                 { wait_async_0(); }
#pragma unroll
        for (int tt = 0; tt < TSTEP; ++tt) {
            float kk[VEC], vv[VEC], nf4[VEC], df4[VEC];
            splat(*(const float4*)&sk[buf][tt][tid * VEC], kk);
            splat(*(const float4*)&sv[buf][tt][tid * VEC], vv);
#pragma unroll
            for (int i = 0; i < VEC; ++i) {
                const float ek  = __expf(fminf(kk[i], 30.0f));
                const float ekv = ek * vv[i];
                aF[i] = aF[i] * dec[i] + ekv;        // forward local scan
                bF[i] = bF[i] * dec[i] + ek;
                aB[i] += pw[i] * ekv;                // backward aggregate
                bB[i] += pw[i] * ek;
                pw[i] *= dec[i];
                nf4[i] = aF[i]; df4[i] = bF[i];
            }
            const size_t o = base + (size_t)(it * TSTEP + tt) * CC;
            *(float4*)(nfl + o) = pack(nf4);
            *(float4*)(dfl + o) = pack(df4);
        }
    }
    const size_t ao = ((size_t)b * NCHUNK + ch) * (size_t)CC + c0;
    *(float4*)(aggFa + ao) = pack(aF);
    *(float4*)(aggFb + ao) = pack(bF);
    *(float4*)(aggBa + ao) = pack(aB);
    *(float4*)(aggBb + ao) = pack(bB);
}

// ---------------------------------------------------------------------------
// P2: per-channel chunk-carry prefix (forward) and suffix (backward) combine.
// carryF(ch) = state entering chunk ch from the left; carryB(ch) from right.
// decay^L computed as exp(L * log(decay)) for stability.
// ---------------------------------------------------------------------------
__global__ __launch_bounds__(256) void
BiRWKV_chunk_carry(const float* __restrict__ w, const float* __restrict__ tf,
                   const float* __restrict__ aggFa, const float* __restrict__ aggFb,
                   const float* __restrict__ aggBa, const float* __restrict__ aggBb,
                   float* __restrict__ cfA, float* __restrict__ cfB,
                   float* __restrict__ cbA, float* __restrict__ cbB) {
    const int gid  = blockIdx.x * 256 + threadIdx.x;
    const int cpb4 = CC / VEC;
    const int b  = gid / cpb4;
    const int c0 = (gid % cpb4) * VEC;
    const float sfac = 0.5f + 0.5f * tf[b];
    const float logs = __logf(sfac);
    float wv[VEC];
    splat(ld4(w + c0), wv);
    float dL[VEC];
#pragma unroll
    for (int i = 0; i < VEC; ++i)
        dL[i] = __expf((float)LCH * (logs - __expf(wv[i])));  // decay^L

    float ca[VEC] = {0.f, 0.f, 0.f, 0.f}, cb_[VEC] = {0.f, 0.f, 0.f, 0.f};
    for (int ch = 0; ch < NCHUNK; ++ch) {
        const size_t o = ((size_t)b * NCHUNK + ch) * (size_t)CC + c0;
        *(float4*)(cfA + o) = pack(ca);
        *(float4*)(cfB + o) = pack(cb_);
        float ga[VEC], gb[VEC];
        splat(ld4(aggFa + o), ga);
        splat(ld4(aggFb + o), gb);
#pragma unroll
        for (int i = 0; i < VEC; ++i) {
            ca[i]  = ca[i]  * dL[i] + ga[i];
            cb_[i] = cb_[i] * dL[i] + gb[i];
        }
    }
#pragma unroll
    for (int i = 0; i < VEC; ++i) { ca[i] = 0.f; cb_[i] = 0.f; }
    for (int ch = NCHUNK - 1; ch >= 0; --ch) {
        const size_t o = ((size_t)b * NCHUNK + ch) * (size_t)CC + c0;
        *(float4*)(cbA + o) = pack(ca);
        *(float4*)(cbB + o) = pack(cb_);
        float ga[VEC], gb[VEC];
        splat(ld4(aggBa + o), ga);
        splat(ld4(aggBb + o), gb);
#pragma unroll
        for (int i = 0; i < VEC; ++i) {
            ca[i]  = ca[i]  * dL[i] + ga[i];
            cb_[i] = cb_[i] * dL[i] + gb[i];
        }
    }
}

// ---------------------------------------------------------------------------
// P3: backward scan (on the fly from carryB) + forward fix-up + combine + out.
// Iterates each chunk in DESCENDING t so nb/db are a plain recurrence; the
// forward fix-up power decay^(i+1) is exp((i+1)*log decay) (stable).
// k/v streamed through LDS with async-to-LDS double buffering.
// ---------------------------------------------------------------------------
__global__ __launch_bounds__(BLOCK) void
BiRWKVLLADAModel_10222022165143_kernel(
    const float* __restrict__ r, const float* __restrict__ k,
    const float* __restrict__ v, const float* __restrict__ w,
    const float* __restrict__ u, const float* __restrict__ tf,
    const float* __restrict__ nfl, const float* __restrict__ dfl,
    const float* __restrict__ cfA, const float* __restrict__ cfB,
    const float* __restrict__ cbA, const float* __restrict__ cbB,
    float* __restrict__ out) {
    __shared__ __align__(16) float sk[2][TSTEP][CPB];
    __shared__ __align__(16) float sv[2][TSTEP][CPB];
    const int tid = threadIdx.x;
    const int c0  = blockIdx.x * CPB + tid * VEC;
    const int ch  = blockIdx.y;
    const int b   = blockIdx.z;
    const int t0  = ch * LCH;
    const size_t base = ((size_t)b * TT + t0) * (size_t)CC + c0;

    const float tfb  = tf[b];
    const float sfac = 0.5f + 0.5f * tfb;
    const float gain = 0.8f + 0.2f * tfb;
    const float logs = __logf(sfac);
    float wv[VEC], uv[VEC];
    splat(ld4(w + c0), wv);
    splat(ld4(u + c0), uv);
    float ldc[VEC], dec[VEC];
#pragma unroll
    for (int i = 0; i < VEC; ++i) {
        ldc[i] = logs - __expf(wv[i]);   // log(decay)
        dec[i] = __expf(ldc[i]);
    }

    const size_t ao = ((size_t)b * NCHUNK + ch) * (size_t)CC + c0;
    float cfa[VEC], cfb[VEC], nb[VEC], db[VEC];
    splat(ld4(cfA + ao), cfa);
    splat(ld4(cfB + ao), cfb);
    splat(ld4(cbA + ao), nb);   // backward state entering chunk from the right
    splat(ld4(cbB + ao), db);

    const uint32_t lk0 = lds_addr(&sk[0][0][tid * VEC]);
    const uint32_t lv0 = lds_addr(&sv[0][0][tid * VEC]);
    auto prefetch = [&](int it, int buf) {
#pragma unroll
        for (int tt = 0; tt < TSTEP; ++tt) {
            const size_t  o  = base + (size_t)(it * TSTEP + tt) * CC;
            const uint32_t lo = (uint32_t)((buf * TSTEP + tt) * CPB * 4);
            async_copy16(lk0 + lo, k + o);
            async_copy16(lv0 + lo, v + o);
        }
    };

    prefetch(NTILES - 1, 0);
    for (int it = NTILES - 1; it >= 0; --it) {
        const int buf = (NTILES - 1 - it) & 1;
        if (it > 0) { prefetch(it - 1, buf ^ 1); wait_async_le8(); }
        else        { wait_async_0(); }
#pragma unroll
        for (int tt = TSTEP - 1; tt >= 0; --tt) {
            const int    ii = it * TSTEP + tt;      // index within chunk
            const size_t o  = base + (size_t)ii * CC;
            float kk[VEC], vv[VEC], rr[VEC], nfv[VEC], dfv[VEC], ov[VEC];
            splat(*(const float4*)&sk[buf][tt][tid * VEC], kk);
            splat(*(const float4*)&sv[buf][tt][tid * VEC], vv);
            splat(ld4(r + o), rr);
            splat(ld4(nfl + o), nfv);
            splat(ld4(dfl + o), dfv);
#pragma unroll
            for (int i = 0; i < VEC; ++i) {
                const float ek  = __expf(fminf(kk[i], 30.0f));
                const float ekv = ek * vv[i];
                nb[i] = nb[i] * dec[i] + ekv;        // global backward scan
                db[i] = db[i] * dec[i] + ek;
                const float p   = __expf((float)(ii + 1) * ldc[i]); // decay^(i+1)
                const float nf  = nfv[i] + p * cfa[i];
                const float df  = dfv[i] + p * cfb[i];
                const float euk = __expf(fminf(uv[i] + kk[i], 30.0f));
                const float num = nf + nb[i] - ekv + euk * vv[i];
                const float den = df + db[i] - ek + euk;
                const float sig = 1.0f / (1.0f + __expf(-rr[i]));
                ov[i] = sig * (num / fmaxf(den, 1e-8f)) * gain;
            }
            *(float4*)(out + o) = pack(ov);
        }
    }
}

// ---------------------------------------------------------------------------
// Host launcher
// ---------------------------------------------------------------------------
extern "C" void kernel_launch(void* const* d_in, const int* in_sizes, int n_in,
                              void* d_out, int out_size, void* d_ws, size_t ws_size,
                              hipStream_t stream) {
    (void)in_sizes; (void)n_in; (void)out_size; (void)ws_size;
    const float* r  = (const float*)d_in[0];
    const float* k  = (const float*)d_in[1];
    const float* v  = (const float*)d_in[2];
    const float* w  = (const float*)d_in[3];
    const float* u  = (const float*)d_in[4];
    const float* te = (const float*)d_in[5];
    float* out = (float*)d_out;

    float* W = (float*)d_ws;
    const size_t NE = (size_t)BB * TT * CC;      // 33.5M
    const size_t NA = (size_t)BB * NCHUNK * CC;  // 0.52M
    float* tfv   = W;            // B floats (padded to 64)
    float* nfl   = W + 64;
    float* dfl   = nfl + NE;
    float* aggFa = dfl + NE;
    float* aggFb = aggFa + NA;
    float* aggBa = aggFb + NA;
    float* aggBb = aggBa + NA;
    float* cfA   = aggBb + NA;
    float* cfB   = cfA + NA;
    float* cbA   = cfB + NA;
    float* cbB   = cbA + NA;

    BiRWKV_timefac<<<BB, 256, 0, stream>>>(te, tfv);

    dim3 g(CC / CPB, NCHUNK, BB);   // (2, 32, 8) blocks of 256 -> 4096 wave32s
    BiRWKV_fwd_scan<<<g, BLOCK, 0, stream>>>(k, v, w, tfv, nfl, dfl,
                                             aggFa, aggFb, aggBa, aggBb);

    BiRWKV_chunk_carry<<<(BB * CC / VEC) / 256, 256, 0, stream>>>(
        w, tfv, aggFa, aggFb, aggBa, aggBb, cfA, cfB, cbA, cbB);

    BiRWKVLLADAModel_10222022165143_kernel<<<g, BLOCK, 0, stream>>>(
        r, k, v, w, u, tfv, nfl, dfl, cfA, cfB, cbA, cbB, out);
}